// GaussianRasterizer_24927990186051
// MI455X (gfx1250) — compile-verified
//
#include <hip/hip_runtime.h>
#include <math.h>

#define N_G 1024
#define IMG_H 192
#define IMG_W 256
#define FXc 220.0f
#define FYc 220.0f
#define CXc 128.0f
#define CYc 96.0f
#define NEARP 0.01f
#define FARP 10.0f
#define ALPHA_THR 1e-5f
#define CHUNK 64
#define NCHUNK (N_G / CHUNK)

typedef __attribute__((ext_vector_type(2))) float v2f;
typedef __attribute__((ext_vector_type(8))) float v8f;

#if defined(__gfx1250__) && __has_builtin(__builtin_amdgcn_global_load_async_to_lds_b32)
#define HAVE_ASYNC_LDS 1
#else
#define HAVE_ASYNC_LDS 0
#endif

// ws float layout: [0 .. 10*N_G) unsorted SoA, [10*N_G .. 20*N_G) depth-sorted SoA
// array order: 0:u 1:v 2:A 3:B 4:C 5:op 6:z 7:r 8:g 9:b

__global__ void gs_prep(const float* __restrict__ pos, const float* __restrict__ rot,
                        const float* __restrict__ scl, const float* __restrict__ opa,
                        const float* __restrict__ sh,  const float* __restrict__ Twc,
                        float* __restrict__ ws) {
  int i = blockIdx.x * blockDim.x + threadIdx.x;
  if (i >= N_G) return;
  float p0 = pos[3*i], p1 = pos[3*i+1], p2 = pos[3*i+2];
  float xc = Twc[0]*p0 + Twc[1]*p1 + Twc[2]*p2  + Twc[3];
  float yc = Twc[4]*p0 + Twc[5]*p1 + Twc[6]*p2  + Twc[7];
  float zc = Twc[8]*p0 + Twc[9]*p1 + Twc[10]*p2 + Twc[11];
  float az = fabsf(zc);
  const float xfac = fmaxf(CXc/FXc, (IMG_W-CXc)/FXc) * 1.1f;
  const float yfac = fmaxf(CYc/FYc, (IMG_H-CYc)/FYc) * 1.1f;
  bool valid = (az > NEARP) && (az < FARP) &&
               (fabsf(xc) < az*xfac) && (fabsf(yc) < az*yfac);
  float zs = (az < 1e-8f) ? 1e-8f : zc;
  float u = FXc*xc/zs + CXc;
  float v = FYc*yc/zs + CYc;
  // quaternion -> rotation
  float qw = rot[4*i], qx = rot[4*i+1], qy = rot[4*i+2], qz = rot[4*i+3];
  float qinv = 1.0f / (sqrtf(qw*qw + qx*qx + qy*qy + qz*qz) + 1e-12f);
  qw *= qinv; qx *= qinv; qy *= qinv; qz *= qinv;
  float R00 = 1.f-2.f*(qy*qy+qz*qz), R01 = 2.f*(qx*qy-qw*qz), R02 = 2.f*(qx*qz+qw*qy);
  float R10 = 2.f*(qx*qy+qw*qz), R11 = 1.f-2.f*(qx*qx+qz*qz), R12 = 2.f*(qy*qz-qw*qx);
  float R20 = 2.f*(qx*qz-qw*qy), R21 = 2.f*(qy*qz+qw*qx), R22 = 1.f-2.f*(qx*qx+qy*qy);
  float s0 = expf(scl[3*i]), s1 = expf(scl[3*i+1]), s2 = expf(scl[3*i+2]);
  float M00=R00*s0, M01=R01*s1, M02=R02*s2;
  float M10=R10*s0, M11=R11*s1, M12=R12*s2;
  float M20=R20*s0, M21=R21*s1, M22=R22*s2;
  // cov3d = M M^T (symmetric)
  float c00 = M00*M00 + M01*M01 + M02*M02;
  float c01 = M00*M10 + M01*M11 + M02*M12;
  float c02 = M00*M20 + M01*M21 + M02*M22;
  float c11 = M10*M10 + M11*M11 + M12*M12;
  float c12 = M10*M20 + M11*M21 + M12*M22;
  float c22 = M20*M20 + M21*M21 + M22*M22;
  float izs = 1.0f / zs;
  float J00 = FXc*izs, J02 = -FXc*xc*izs*izs;
  float J11 = FYc*izs, J12 = -FYc*yc*izs*izs;
  float t00 = J00*c00 + J02*c02;
  float t01 = J00*c01 + J02*c12;
  float t02 = J00*c02 + J02*c22;
  float t11 = J11*c11 + J12*c12;
  float t12 = J11*c12 + J12*c22;
  float a  = t00*J00 + t02*J02 + 0.3f;
  float b  = t01*J11 + t02*J12;          // cov2d[0,1]
  float cc = t11*J11 + t12*J12 + 0.3f;
  float det  = fmaxf(a*cc - b*b, 1e-12f);
  float idet = 1.0f / det;
  float conA =  cc*idet;
  float conB =  -b*idet;
  float conC =   a*idet;
  float op = opa[i] * (valid ? 1.0f : 0.0f);
  float nrm = sqrtf(xc*xc + yc*yc + zc*zc) + 1e-12f;
  float dx = xc/nrm, dy = yc/nrm, dz = zc/nrm;
  float xx=dx*dx, yy=dy*dy, zz=dz*dz;
  float xy=dx*dy, yz=dy*dz, xz=dx*dz;
  const float C0 = 0.28209479177387814f;
  const float C1 = 0.4886025119029199f;
  const float C2_0 = 1.0925484305920792f, C2_1 = -1.0925484305920792f,
              C2_2 = 0.31539156525252005f, C2_3 = -1.0925484305920792f,
              C2_4 = 0.5462742152960396f;
  const float C3_0 = -0.5900435899266435f, C3_1 = 2.890611442640554f,
              C3_2 = -0.4570457994644658f, C3_3 = 0.3731763325901154f,
              C3_4 = -0.4570457994644658f, C3_5 = 1.445305721320277f,
              C3_6 = -0.5900435899266435f;
  const float* S = sh + i*48;
  float col[3];
  #pragma unroll
  for (int ch = 0; ch < 3; ++ch) {
    float r = C0*S[0*3+ch] - C1*dy*S[1*3+ch] + C1*dz*S[2*3+ch] - C1*dx*S[3*3+ch];
    r += C2_0*xy*S[4*3+ch] + C2_1*yz*S[5*3+ch] + C2_2*(2.f*zz-xx-yy)*S[6*3+ch]
       + C2_3*xz*S[7*3+ch] + C2_4*(xx-yy)*S[8*3+ch];
    r += C3_0*dy*(3.f*xx-yy)*S[9*3+ch] + C3_1*xy*dz*S[10*3+ch]
       + C3_2*dy*(4.f*zz-xx-yy)*S[11*3+ch]
       + C3_3*dz*(2.f*zz-3.f*xx-3.f*yy)*S[12*3+ch]
       + C3_4*dx*(4.f*zz-xx-yy)*S[13*3+ch]
       + C3_5*dz*(xx-yy)*S[14*3+ch] + C3_6*dx*(xx-3.f*yy)*S[15*3+ch];
    col[ch] = fminf(fmaxf(r + 0.5f, 0.0f), 1.0f);
  }
  ws[0*N_G+i]=u;      ws[1*N_G+i]=v;      ws[2*N_G+i]=conA;  ws[3*N_G+i]=conB;
  ws[4*N_G+i]=conC;   ws[5*N_G+i]=op;     ws[6*N_G+i]=zc;
  ws[7*N_G+i]=col[0]; ws[8*N_G+i]=col[1]; ws[9*N_G+i]=col[2];
}

// rank sort ascending z; tie-break (j > i) reproduces reference argsort(-z) order
// processed front-to-back.
__global__ void gs_sort(float* __restrict__ ws) {
  int i = blockIdx.x * blockDim.x + threadIdx.x;
  if (i >= N_G) return;
  const float* z = ws + 6*N_G;
  float zi = z[i];
  int rank = 0;
  for (int j = 0; j < N_G; ++j) {
    float zj = z[j];
    rank += ((zj < zi) || (zj == zi && j > i)) ? 1 : 0;
  }
  float* dst = ws + 10*N_G;
  #pragma unroll
  for (int a = 0; a < 10; ++a) dst[a*N_G + rank] = ws[a*N_G + i];
}

// ---- async global -> LDS staging helpers -----------------------------------

typedef __attribute__((address_space(1))) int g_i32;
typedef __attribute__((address_space(3))) int l_i32;

__device__ __forceinline__ void copy32_to_lds(const float* gsrc, float* ldst) {
#if HAVE_ASYNC_LDS
  __builtin_amdgcn_global_load_async_to_lds_b32(
      (g_i32*)(uintptr_t)gsrc,
      (l_i32*)(uintptr_t)ldst,
      0, 0);
#else
  *ldst = *gsrc;
#endif
}

__device__ __forceinline__ void async_wait_le10() {
#if HAVE_ASYNC_LDS
# if __has_builtin(__builtin_amdgcn_s_wait_asynccnt)
  __builtin_amdgcn_s_wait_asynccnt(10);
# else
  asm volatile("s_wait_asynccnt 10" ::: "memory");
# endif
#endif
}

__device__ __forceinline__ void async_wait_0() {
#if HAVE_ASYNC_LDS
# if __has_builtin(__builtin_amdgcn_s_wait_asynccnt)
  __builtin_amdgcn_s_wait_asynccnt(0);
# else
  asm volatile("s_wait_asynccnt 0" ::: "memory");
# endif
#endif
}

// LDS chunk buffer layout (floats): u[64] v[64] A[64] B[64] C[64] op[64] col4[64*4]
#define UO 0
#define VO 64
#define AO 128
#define BO 192
#define CO 256
#define OO 320
#define C4 384
#define BUF_F (C4 + CHUNK*4)   // 640 floats per buffer

__device__ __forceinline__ void stage_chunk(float* D, const float* su, int c, int t) {
  int g = c*CHUNK + t;
  copy32_to_lds(&su[0*N_G+g], &D[UO+t]);
  copy32_to_lds(&su[1*N_G+g], &D[VO+t]);
  copy32_to_lds(&su[2*N_G+g], &D[AO+t]);
  copy32_to_lds(&su[3*N_G+g], &D[BO+t]);
  copy32_to_lds(&su[4*N_G+g], &D[CO+t]);
  copy32_to_lds(&su[5*N_G+g], &D[OO+t]);
  copy32_to_lds(&su[7*N_G+g], &D[C4+t*4+0]);   // r
  copy32_to_lds(&su[8*N_G+g], &D[C4+t*4+1]);   // g
  copy32_to_lds(&su[9*N_G+g], &D[C4+t*4+2]);   // b
  copy32_to_lds(&su[6*N_G+g], &D[C4+t*4+3]);   // depth
}

__global__ void __launch_bounds__(256) gs_raster(const float* __restrict__ ws,
                                                 float* __restrict__ out) {
  __shared__ float lbuf[2][BUF_F];  // double-buffered gaussian chunk
  __shared__ float outs[8*32*4];    // per-wave: [pixel][channel]

  const float* su = ws + 10*N_G;
  int tid  = threadIdx.x;
  int pid  = blockIdx.x * 256 + tid;
  int lane = tid & 31;
  int wv   = tid >> 5;
  int laneN = lane & 15;
  bool loHalf = lane < 16;
  float px = (float)(pid % IMG_W) + 0.5f;
  float py = (float)(pid / IMG_W) + 0.5f;

  float Tcur = 1.0f;
  v8f accLo = {0.f,0.f,0.f,0.f,0.f,0.f,0.f,0.f};
  v8f accHi = {0.f,0.f,0.f,0.f,0.f,0.f,0.f,0.f};

  // prologue: async-stage chunk 0 into buffer 0 (loader = waves 0,1 fully active)
  if (tid < CHUNK) stage_chunk(&lbuf[0][0], su, 0, tid);

  for (int c = 0; c < NCHUNK; ++c) {
    int cur = c & 1;
    if (c + 1 < NCHUNK) {
      // issue next chunk's async copies, then wait until only those 10 remain:
      // in-order completion guarantees the current chunk has fully landed.
      if (tid < CHUNK) stage_chunk(&lbuf[cur ^ 1][0], su, c + 1, tid);
      async_wait_le10();
    } else {
      async_wait_0();
    }
    __syncthreads();

    float* Lb = &lbuf[cur][0];
    for (int grp = 0; grp < CHUNK/4; ++grp) {
      float w4[4];
      #pragma unroll
      for (int k = 0; k < 4; ++k) {
        int idx = grp*4 + k;
        float ddx = px - Lb[UO+idx];
        float ddy = py - Lb[VO+idx];
        float pw = -0.5f*(Lb[AO+idx]*ddx*ddx + Lb[CO+idx]*ddy*ddy)
                 - Lb[BO+idx]*ddx*ddy;
        float alpha = fminf(0.99f, Lb[OO+idx] * __expf(fminf(pw, 0.0f)));
        alpha = (alpha < ALPHA_THR) ? 0.0f : alpha;
        w4[k] = alpha * Tcur;
        Tcur *= (1.0f - alpha);
      }
      // A matrices per ISA layout: lanes 0-15 hold K=0..1, lanes 16-31 hold K=2..3.
      float s2 = __shfl(w4[2], laneN, 32);        // from lane L-16 for upper half
      float s3 = __shfl(w4[3], laneN, 32);
      float g0 = __shfl(w4[0], 16 + laneN, 32);   // from lane L+16 for lower half
      float g1 = __shfl(w4[1], 16 + laneN, 32);
      v2f aLo, aHi, bM;
      aLo.x = loHalf ? w4[0] : s2;
      aLo.y = loHalf ? w4[1] : s3;
      aHi.x = loHalf ? g0 : w4[2];
      aHi.y = loHalf ? g1 : w4[3];
      // B: 4x16, row k = gaussian grp*4+k, col n = channel (r,g,b,z,0...).
      int gi = grp*4 + (loHalf ? 0 : 2);
      bM.x = (laneN < 4) ? Lb[C4 + gi*4 + laneN]     : 0.0f;
      bM.y = (laneN < 4) ? Lb[C4 + (gi+1)*4 + laneN] : 0.0f;
      accLo = __builtin_amdgcn_wmma_f32_16x16x4_f32(false, aLo, false, bM,
                                                    (short)0, accLo, false, false);
      accHi = __builtin_amdgcn_wmma_f32_16x16x4_f32(false, aHi, false, bM,
                                                    (short)0, accHi, false, false);
    }
    __syncthreads();   // buffer 'cur' will be re-staged next iteration
  }

  // Scatter accumulator (C/D layout: VGPR r, lanes0-15 -> M=r, lanes16-31 -> M=r+8, N=lane%16)
  if (laneN < 4) {
    int base = wv*128;
    #pragma unroll
    for (int r = 0; r < 8; ++r) {
      int m = r + (loHalf ? 0 : 8);
      outs[base + m*4 + laneN]        = accLo[r];
      outs[base + (16 + m)*4 + laneN] = accHi[r];
    }
  }
  __syncthreads();
  int base = wv*128 + lane*4;
  out[pid*3+0] = outs[base+0];
  out[pid*3+1] = outs[base+1];
  out[pid*3+2] = outs[base+2];
  out[IMG_H*IMG_W*3 + pid] = outs[base+3];
}

extern "C" void kernel_launch(void* const* d_in, const int* in_sizes, int n_in,
                              void* d_out, int out_size, void* d_ws, size_t ws_size,
                              hipStream_t stream) {
  (void)in_sizes; (void)n_in; (void)out_size; (void)ws_size;
  const float* pos = (const float*)d_in[0];
  const float* rot = (const float*)d_in[1];
  const float* scl = (const float*)d_in[2];
  const float* opa = (const float*)d_in[3];
  const float* sh  = (const float*)d_in[4];
  const float* twc = (const float*)d_in[5];
  float* ws  = (float*)d_ws;
  float* out = (float*)d_out;
  gs_prep<<<N_G/256, 256, 0, stream>>>(pos, rot, scl, opa, sh, twc, ws);
  gs_sort<<<N_G/256, 256, 0, stream>>>(ws);
  gs_raster<<<(IMG_H*IMG_W)/256, 256, 0, stream>>>(ws, out);
}